// CRF_9715216024088
// MI455X (gfx1250) — compile-verified
//
#include <hip/hip_runtime.h>
#include <hip/hip_bf16.h>
#include <stdint.h>

#define Bv 128
#define Tv 1024
#define Kv 128

// ---------------------------------------------------------------------------
// Forward Viterbi: one block per batch, thread j owns state j.
//  - trans column j held in 128 VGPRs per thread
//  - alpha double-buffered in LDS (one barrier per step)
//  - logits prefetched 3 steps ahead via gfx1250 async-to-LDS (4-slot ring),
//    retired with s_wait_asynccnt <= 2 (async loads complete in order)
//  - backpointers (u8) streamed to global (front of d_out)
// ---------------------------------------------------------------------------
__global__ void __launch_bounds__(128)
crf_fwd(const float* __restrict__ logits,     // [B,T,K]
        const float* __restrict__ trans,      // [K,K]
        const int*   __restrict__ slen,       // [B]
        unsigned char* __restrict__ bp,       // [B,T,K] u8 (t=0 row unused)
        int*   __restrict__ tags,             // [B,T] (we write last slot)
        float* __restrict__ score)            // [B]
{
    __shared__ __align__(16) float abuf[2][Kv];   // alpha double buffer
    __shared__ __align__(16) float lbuf[4][Kv];   // logit ring (async target)

    const int j   = threadIdx.x;
    const int b   = blockIdx.x;
    const int len = slen[b];
    const float* lg = logits + (size_t)b * Tv * Kv;

    // transitions column j -> registers (coalesced across lanes per row)
    float tc[Kv];
#pragma unroll
    for (int i = 0; i < Kv; ++i) tc[i] = trans[i * Kv + j];

    float myA = lg[j];          // alpha_0[j] = logits[b,0,j]
    abuf[0][j] = myA;

    // prologue: issue async loads for steps 1..3 (dummy safe address past end)
#pragma unroll
    for (int s = 1; s <= 3; ++s) {
        unsigned laddr = (unsigned)(unsigned long long)&lbuf[s & 3][j];
        const float* g = (s < len) ? (lg + (size_t)s * Kv + j) : (lg + j);
        asm volatile("global_load_async_to_lds_b32 %0, %1, off"
                     :: "v"(laddr), "v"(g) : "memory");
    }
    __syncthreads();

    unsigned char* bpp  = bp + ((size_t)b * Tv + 1) * Kv + j;  // bp[b,1,j]
    const float*   gpre = lg + (size_t)4 * Kv + j;             // step t+3 at t=1

    for (int t = 1; t < len; ++t) {
        const int wa = t & 1;        // write alpha buffer
        const int pa = wa ^ 1;       // previous alpha buffer

        // 3 loads in flight; in-order completion => cnt<=2 means step t landed
        asm volatile("s_wait_asynccnt 2" ::: "memory");

        // keep the ring full: issue step t+3 (dummy past end of sequence);
        // slot (t+3)&3 == (t-1)&3 was consumed last iteration -> safe.
        {
            unsigned laddr = (unsigned)(unsigned long long)&lbuf[(t + 3) & 3][j];
            const float* g = (t + 3 < len) ? gpre : (lg + j);
            asm volatile("global_load_async_to_lds_b32 %0, %1, off"
                         :: "v"(laddr), "v"(g) : "memory");
            gpre += Kv;
        }

        const float logit = lbuf[t & 3][j];

        // 4 segment-ordered accumulators (ILP) preserving first-argmax ties
        float bv0 = -3.402823466e38f, bv1 = bv0, bv2 = bv0, bv3 = bv0;
        int   bi0 = 0, bi1 = 32, bi2 = 64, bi3 = 96;
#pragma unroll
        for (int c = 0; c < Kv / 4; ++c) {
            const float4 a = *(const float4*)&abuf[pa][4 * c];
            const float s0 = a.x + tc[4 * c + 0];
            const float s1 = a.y + tc[4 * c + 1];
            const float s2 = a.z + tc[4 * c + 2];
            const float s3 = a.w + tc[4 * c + 3];
            if (c < 8) {
                if (s0 > bv0) { bv0 = s0; bi0 = 4 * c + 0; }
                if (s1 > bv0) { bv0 = s1; bi0 = 4 * c + 1; }
                if (s2 > bv0) { bv0 = s2; bi0 = 4 * c + 2; }
                if (s3 > bv0) { bv0 = s3; bi0 = 4 * c + 3; }
            } else if (c < 16) {
                if (s0 > bv1) { bv1 = s0; bi1 = 4 * c + 0; }
                if (s1 > bv1) { bv1 = s1; bi1 = 4 * c + 1; }
                if (s2 > bv1) { bv1 = s2; bi1 = 4 * c + 2; }
                if (s3 > bv1) { bv1 = s3; bi1 = 4 * c + 3; }
            } else if (c < 24) {
                if (s0 > bv2) { bv2 = s0; bi2 = 4 * c + 0; }
                if (s1 > bv2) { bv2 = s1; bi2 = 4 * c + 1; }
                if (s2 > bv2) { bv2 = s2; bi2 = 4 * c + 2; }
                if (s3 > bv2) { bv2 = s3; bi2 = 4 * c + 3; }
            } else {
                if (s0 > bv3) { bv3 = s0; bi3 = 4 * c + 0; }
                if (s1 > bv3) { bv3 = s1; bi3 = 4 * c + 1; }
                if (s2 > bv3) { bv3 = s2; bi3 = 4 * c + 2; }
                if (s3 > bv3) { bv3 = s3; bi3 = 4 * c + 3; }
            }
        }
        // combine in segment order (strict > keeps lowest index on ties)
        float best = bv0; int bi = bi0;
        if (bv1 > best) { best = bv1; bi = bi1; }
        if (bv2 > best) { best = bv2; bi = bi2; }
        if (bv3 > best) { best = bv3; bi = bi3; }

        myA = best + logit;
        abuf[wa][j] = myA;
        *bpp = (unsigned char)bi;
        bpp += Kv;
        __syncthreads();
    }

    // padding region: identity backpointers, written as dword patterns.
    // 128 threads = 4 groups of 32 lanes; group g covers row t+g, lane d
    // writes bytes 4d..4d+3 of the identity pattern.
    {
        const int g = j >> 5, d = j & 31;
        const unsigned pat = (unsigned)(4 * d) | ((unsigned)(4 * d + 1) << 8) |
                             ((unsigned)(4 * d + 2) << 16) | ((unsigned)(4 * d + 3) << 24);
        unsigned* bpd = (unsigned*)(bp + (size_t)b * Tv * Kv);
        for (int t = len + g; t < Tv; t += 4)
            bpd[t * 32 + d] = pat;
    }

    // best_score / last_tag over final alpha (first-argmax semantics)
    if (j == 0) {
        const float* fa = abuf[(len - 1) & 1];
        float bs = fa[0]; int bt = 0;
        for (int i = 1; i < Kv; ++i) {
            const float v = fa[i];
            if (v > bs) { bs = v; bt = i; }
        }
        score[b]              = bs;
        tags[b * Tv + Tv - 1] = bt;
    }
}

// ---------------------------------------------------------------------------
// Backtrace: one block per batch. Stage this batch's 128 KB of backpointers
// into LDS (CDNA5 has 320 KB/WGP), then thread 0 pointer-chases at LDS
// latency instead of L2 latency. Writes tags[b, 0..T-2].
// ---------------------------------------------------------------------------
__global__ void __launch_bounds__(128)
crf_backtrace(const unsigned char* __restrict__ bp,   // [B,T,K] u8
              int* __restrict__ tags)                 // [B,T]
{
    extern __shared__ unsigned char sbp[];            // Tv*Kv = 128 KB
    const int b   = blockIdx.x;
    const int tid = threadIdx.x;

    const uint4* src = (const uint4*)(bp + (size_t)b * Tv * Kv);
    uint4*       dst = (uint4*)sbp;
    const int n16 = Tv * Kv / 16;                     // 8192 vec16 loads
    for (int i = tid; i < n16; i += 128) dst[i] = src[i];
    __syncthreads();

    if (tid == 0) {
        int tag = tags[b * Tv + Tv - 1];
        for (int t = Tv - 1; t >= 1; --t) {
            tag = (int)sbp[t * Kv + tag];
            tags[b * Tv + t - 1] = tag;
        }
    }
}

// ---------------------------------------------------------------------------
// One-hot expansion: one wave32 per (b,t) row; lane l writes float4 at 4l.
// Overwrites the backpointer scratch region of d_out with the real output.
// ---------------------------------------------------------------------------
__global__ void __launch_bounds__(256)
crf_onehot(const int* __restrict__ tags,   // [B,T]
           float* __restrict__ out)        // [B,T,K]
{
    const int wave = threadIdx.x >> 5;               // 0..7
    const int lane = threadIdx.x & 31;
    const int row  = blockIdx.x * 8 + wave;          // b*T + t
    const int tag  = tags[row];
    const int base = lane * 4;
    float4 v;
    v.x = (base + 0 == tag) ? 1.0f : 0.0f;
    v.y = (base + 1 == tag) ? 1.0f : 0.0f;
    v.z = (base + 2 == tag) ? 1.0f : 0.0f;
    v.w = (base + 3 == tag) ? 1.0f : 0.0f;
    *(float4*)(out + (size_t)row * Kv + base) = v;
}

extern "C" void kernel_launch(void* const* d_in, const int* in_sizes, int n_in,
                              void* d_out, int out_size, void* d_ws, size_t ws_size,
                              hipStream_t stream) {
    const float* logits = (const float*)d_in[0];   // [B,T,K] f32
    const float* trans  = (const float*)d_in[1];   // [K,K]   f32
    const int*   slen   = (const int*)d_in[2];     // [B]     i32

    float* out   = (float*)d_out;                          // onehot [B,T,K] + score [B]
    float* score = out + (size_t)Bv * Tv * Kv;             // tail of d_out
    unsigned char* bpbuf = (unsigned char*)d_out;          // reuse front 16 MB as u8 bp scratch
    int* tags = (int*)d_ws;                                // [B,T] i32 in workspace (512 KB)

    crf_fwd      <<<Bv,            128, 0,          stream>>>(logits, trans, slen, bpbuf, tags, score);
    crf_backtrace<<<Bv,            128, Tv * Kv,    stream>>>(bpbuf, tags);
    crf_onehot   <<<(Bv * Tv) / 8, 256, 0,          stream>>>(tags, out);
}